// BandpassFilter_54614804136149
// MI455X (gfx1250) — compile-verified
//
#include <hip/hip_runtime.h>

// Bandpass biquad (lowpass - highpass) on [64][220500] f32.
// Block-parallel exact IIR: per-chunk zero-state pass (K1), 2x2 state scan (K2),
// final pass with correct incoming state fused with low-high subtraction (K3).
// 16 time steps advanced per step via V_WMMA_F32_16X16X4_F32:
//   Y(16t x 16ch) = G(16x20 Toeplitz of full biquad impulse response) @ Xwin(20x16)
//                 + P(cols 2,3 = p1,p2) @ S(rows 2,3 = y[-1],y[-2])
// The FIR part is folded into G, so B operands are raw x from LDS (shared by both
// filters). Carry rows (K=2,3) live in lanes 16-31 = exactly where acc[7]/acc[6]
// (M=15/14) sit, so the inter-tile carry needs no cross-lane shuffle at all.
// X staging uses double-buffered GLOBAL_LOAD_ASYNC_TO_LDS_B128 (ASYNCcnt) so the
// next sub-block streams into LDS while WMMA crunches the current one. Because
// the filter is causal (G[m][j]==0 for j>m+2), out-of-range staging is handled by
// clamping addresses instead of per-element guards; only chunk 0's 2-sample left
// halo genuinely needs zeros.
// Memory-bound problem (~113MB min traffic -> ~4.8us @23.3TB/s); this design
// moves ~170MB total (~7.3us) and keeps all math in f32.

#define PI_F 3.14159265358979323846f

typedef __attribute__((ext_vector_type(2))) float v2f;
typedef __attribute__((ext_vector_type(8))) float v8f;

constexpr int T_LEN   = 220500;
constexpr int NCH     = 64;                       // 32 batch * 2 channels
constexpr int CHUNK_L = 864;                      // multiple of 16
constexpr int NCHUNK  = (T_LEN + CHUNK_L - 1) / CHUNK_L;   // 256
constexpr int NCHGRP  = NCH / 16;                 // 4 channel groups of 16
constexpr int SB      = 288;                      // LDS sub-block (3 per chunk)
constexpr int XW_F4   = 80;                       // staged width: 80 float4 = 320 f
constexpr int XS_STR  = 324;                      // floats; 324*4 % 16 == 0, bank+4/row
constexpr int YS_STR  = 292;                      // floats; 292*4 % 16 == 0
constexpr int NISSUE  = 16 * 3;                   // async issues per sub-block (48)

struct FiltParams {
  float m00, m01, m10, m11;     // 2x2 state transition over CHUNK_L samples
  float G[16][20];              // Toeplitz of full impulse response, K-padded
  float p1[16], p2[16];         // homogeneous responses to y[-1]=1 / y[-2]=1
};

constexpr size_t PARAMS_BYTES = 8192;
constexpr size_t ZS_FLOATS    = 2ull * NCH * NCHUNK * 2;   // 65536 floats

__device__ __forceinline__ v8f wmma4(v2f a, v2f b, v8f c) {
  // D = A(16x4) * B(4x16) + C   (f32, wave32)
  return __builtin_amdgcn_wmma_f32_16x16x4_f32(
      false, a, false, b, (short)0, c, false, false);
}

__global__ void bq_setup(const int* __restrict__ sr_p,
                         const float* __restrict__ cut_lo,
                         const float* __restrict__ cut_hi,
                         FiltParams* __restrict__ P) {
  int f = threadIdx.x;
  if (f >= 2) return;
  float sr     = (float)sr_p[0];
  float cutoff = (f == 0) ? cut_lo[0] : cut_hi[0];
  float w0 = 2.0f * PI_F * cutoff / sr;
  float c = cosf(w0), s = sinf(w0);
  float alpha = s / (2.0f * 0.707f);
  float a0 = 1.0f + alpha;
  float b0, b1, b2;
  if (f == 0) { b0 = 0.5f * (1.0f - c); b1 = 1.0f - c; }      // lowpass
  else        { b0 = 0.5f * (1.0f + c); b1 = -(1.0f + c); }   // highpass
  b2 = b0;
  float a1 = -2.0f * c, a2 = 1.0f - alpha;
  float inv = 1.0f / a0;
  b0 *= inv; b1 *= inv; b2 *= inv; a1 *= inv; a2 *= inv;

  FiltParams& p = P[f];

  // h: impulse response of the pure recursion; g = b (x) h (full biquad IR).
  float h[18];
  h[0] = 1.0f;
  h[1] = -a1;
  for (int n = 2; n < 18; ++n) h[n] = -a1 * h[n - 1] - a2 * h[n - 2];
  float g[18];
  for (int n = 0; n < 18; ++n) {
    float v = b0 * h[n];
    if (n >= 1) v += b1 * h[n - 1];
    if (n >= 2) v += b2 * h[n - 2];
    g[n] = v;
  }
  // Y[m] = sum_j G[m][j] * xwin[j], xwin[j] = x[t0 - 2 + j], j = 0..17.
  for (int m = 0; m < 16; ++m)
    for (int j = 0; j < 20; ++j) {
      int d = m - j + 2;
      p.G[m][j] = (j < 18 && d >= 0) ? g[d] : 0.0f;
    }

  { float x1 = 1.0f, x2 = 0.0f;
    for (int n = 0; n < 16; ++n) { float y = -a1 * x1 - a2 * x2; p.p1[n] = y; x2 = x1; x1 = y; } }
  { float x1 = 0.0f, x2 = 1.0f;
    for (int n = 0; n < 16; ++n) { float y = -a1 * x1 - a2 * x2; p.p2[n] = y; x2 = x1; x1 = y; } }

  float cm[2][2];
  for (int col = 0; col < 2; ++col) {
    float x1 = (col == 0) ? 1.0f : 0.0f;
    float x2 = (col == 0) ? 0.0f : 1.0f;
    for (int n = 0; n < CHUNK_L; ++n) { float y = -a1 * x1 - a2 * x2; x2 = x1; x1 = y; }
    cm[0][col] = x1; cm[1][col] = x2;
  }
  p.m00 = cm[0][0]; p.m01 = cm[0][1]; p.m10 = cm[1][0]; p.m11 = cm[1][1];
}

__global__ void bq_scan(const FiltParams* __restrict__ P,
                        const float* __restrict__ zs,
                        float* __restrict__ sin_st) {
  int s = threadIdx.x;                   // strip = filter*64 + channel
  if (s >= 2 * NCH) return;
  const FiltParams& p = P[s / NCH];
  float m00 = p.m00, m01 = p.m01, m10 = p.m10, m11 = p.m11;
  float s1 = 0.f, s2 = 0.f;
  for (int i = 0; i < NCHUNK; ++i) {
    size_t base = ((size_t)s * NCHUNK + i) * 2;
    sin_st[base]     = s1;
    sin_st[base + 1] = s2;
    float d1 = zs[base], d2 = zs[base + 1];
    float t1 = m00 * s1 + m01 * s2 + d1;
    float t2 = m10 * s1 + m11 * s2 + d2;
    s1 = t1; s2 = t2;
  }
}

// Async-stage one sub-block of x (16 rows x 320 floats, left edge at t0+sbs-4)
// into LDS buffer `Xb` with GLOBAL_LOAD_ASYNC_TO_LDS_B128. Addresses clamped to
// [0, T-4]; causality makes clamped data harmless (zero G columns / discarded).
__device__ __forceinline__ void stage_async(const float* __restrict__ x,
                                            float* Xb, int ch0, int tg0a, int lane) {
  for (int r = 0; r < 16; ++r) {
    const unsigned long long gbase =
        (unsigned long long)(uintptr_t)(x + (size_t)(ch0 + r) * T_LEN);
    const unsigned lbase = (unsigned)(uintptr_t)(&Xb[r * XS_STR]);
#pragma unroll
    for (int i = 0; i < 3; ++i) {
      int q = i * 32 + lane;
      if (q < XW_F4) {
        int tg = tg0a + 4 * q;
        tg = tg < 0 ? 0 : tg;
        tg = tg > (T_LEN - 4) ? (T_LEN - 4) : tg;
        unsigned long long gp = gbase + (unsigned long long)tg * 4ull;
        unsigned loff = lbase + (unsigned)(q * 16);
        asm volatile("global_load_async_to_lds_b128 %0, %1, off"
                     :: "v"(loff), "v"(gp) : "memory");
      }
    }
  }
}

template <bool WRITE_OUT>
__global__ void bq_chunks(const float* __restrict__ x, float* __restrict__ out,
                          const FiltParams* __restrict__ P,
                          float* __restrict__ zs,
                          const float* __restrict__ sin_st) {
  __shared__ alignas(16) float Xs[2 * 16 * XS_STR];          // double-buffered
  __shared__ alignas(16) float Ys[WRITE_OUT ? 16 * YS_STR : 1];

  const int lane = threadIdx.x;          // one wave32 per workgroup
  const int n    = lane & 15;
  const bool lo  = lane < 16;
  const int ch0  = blockIdx.x * 16;
  const int chunk = blockIdx.y;
  const int t0    = chunk * CHUNK_L;
  const int chunk_len = min(CHUNK_L, T_LEN - t0);

  // A fragments: 5 K-groups of G, plus P with p1,p2 in columns 2,3 (high half).
  v2f Afrag[2][5];
  v2f Pfrag[2];
  float prev7[2], prev6[2];              // own-lane y[t-1], y[t-2] (high half valid)
  const int kofs = lo ? 0 : 2;
#pragma unroll
  for (int f = 0; f < 2; ++f) {
    const FiltParams& p = P[f];
#pragma unroll
    for (int g = 0; g < 5; ++g) {
      Afrag[f][g].x = p.G[n][4 * g + kofs];
      Afrag[f][g].y = p.G[n][4 * g + kofs + 1];
    }
    Pfrag[f].x = lo ? 0.0f : p.p1[n];    // A col 2
    Pfrag[f].y = lo ? 0.0f : p.p2[n];    // A col 3
    if constexpr (WRITE_OUT) {
      size_t base = (((size_t)f * NCH + ch0 + n) * NCHUNK + chunk) * 2;
      prev7[f] = sin_st[base];           // y[-1]
      prev6[f] = sin_st[base + 1];       // y[-2]
    } else {
      prev7[f] = 0.0f; prev6[f] = 0.0f;
    }
  }

  const int nsb = (chunk_len + SB - 1) / SB;
  stage_async(x, &Xs[0], ch0, t0 - 4, lane);                  // prefetch sb 0

  for (int sb = 0; sb < nsb; ++sb) {
    const int sbs    = sb * SB;
    const int sb_len = min(SB, chunk_len - sbs);

    if (sb + 1 < nsb) {
      stage_async(x, &Xs[((sb + 1) & 1) * 16 * XS_STR], ch0, t0 + sbs + SB - 4, lane);
      // Loads complete in order: <=48 outstanding => current buffer landed.
      asm volatile("s_wait_asynccnt 48" ::: "memory");
    } else {
      asm volatile("s_wait_asynccnt 0" ::: "memory");
    }

    float* Xb = &Xs[(sb & 1) * 16 * XS_STR];
    if (chunk == 0 && sb == 0 && lo) {   // true zeros for the 2-sample left halo
#pragma unroll
      for (int j = 0; j < 4; ++j) Xb[n * XS_STR + j] = 0.0f;
    }

    const int ntiles = (sb_len + 15) >> 4;
    for (int tt = 0; tt < ntiles; ++tt) {
      const int tl = tt * 16;
      const float* row = &Xb[n * XS_STR + tl + 2];  // row[j] = xwin[j]
      // Raw-x B fragments, shared by both filters.
      v2f Bf[5];
#pragma unroll
      for (int g = 0; g < 5; ++g) {
        const int k0 = 4 * g + kofs;
        Bf[g].x = row[k0];
        Bf[g].y = row[k0 + 1];
      }
      v8f accs[2];
#pragma unroll
      for (int f = 0; f < 2; ++f) {
        // Carry fragment: rows 2,3 (high half) = own-lane previous acc[7]/acc[6].
        v2f Sfrag;
        Sfrag.x = lo ? 0.0f : prev7[f];
        Sfrag.y = lo ? 0.0f : prev6[f];
        v8f acc = {0.f, 0.f, 0.f, 0.f, 0.f, 0.f, 0.f, 0.f};
        acc = wmma4(Pfrag[f], Sfrag, acc);       // state (homogeneous) part
#pragma unroll
        for (int g = 0; g < 5; ++g)              // K=20 as 5x K=4
          acc = wmma4(Afrag[f][g], Bf[g], acc);
        prev7[f] = acc[7];                       // y[15] at (M=15,N=n), lane 16+n
        prev6[f] = acc[6];                       // y[14]
        accs[f] = acc;
      }
      if constexpr (WRITE_OUT) {
        v8f d = accs[0] - accs[1];               // low - high, same layout
        const int tb = tl + (lo ? 0 : 8);
#pragma unroll
        for (int r = 0; r < 8; ++r)
          Ys[n * YS_STR + tb + r] = d[r];
      }
    }

    if constexpr (WRITE_OUT) {
      // Coalesced float4 flush (sb_len is 288 or 180, both /4; rows 16B aligned).
      const int nf4 = sb_len >> 2;
      for (int r = 0; r < 16; ++r) {
        const float4* yr = (const float4*)&Ys[r * YS_STR];
        float4* orow = (float4*)(out + (size_t)(ch0 + r) * T_LEN + t0 + sbs);
        for (int q = lane; q < nf4; q += 32) orow[q] = yr[q];
      }
    }
  }

  if constexpr (!WRITE_OUT) {
    if (!lo) {                                   // high half owns the state
#pragma unroll
      for (int f = 0; f < 2; ++f) {
        size_t base = (((size_t)f * NCH + ch0 + n) * NCHUNK + chunk) * 2;
        zs[base]     = prev7[f];                 // y[L-1]
        zs[base + 1] = prev6[f];                 // y[L-2]
      }
    }
  }
}

extern "C" void kernel_launch(void* const* d_in, const int* in_sizes, int n_in,
                              void* d_out, int out_size, void* d_ws, size_t ws_size,
                              hipStream_t stream) {
  (void)in_sizes; (void)n_in; (void)out_size; (void)ws_size;
  const float* audio = (const float*)d_in[0];
  const int*   sr    = (const int*)d_in[1];
  const float* clo   = (const float*)d_in[2];
  const float* chi   = (const float*)d_in[3];
  float* out = (float*)d_out;

  char* ws = (char*)d_ws;
  FiltParams* P = (FiltParams*)ws;
  float* zs  = (float*)(ws + PARAMS_BYTES);
  float* sst = (float*)(ws + PARAMS_BYTES + ZS_FLOATS * sizeof(float));

  bq_setup<<<1, 64, 0, stream>>>(sr, clo, chi, P);
  bq_chunks<false><<<dim3(NCHGRP, NCHUNK), 32, 0, stream>>>(audio, nullptr, P, zs, nullptr);
  bq_scan<<<1, 128, 0, stream>>>(P, zs, sst);
  bq_chunks<true><<<dim3(NCHGRP, NCHUNK), 32, 0, stream>>>(audio, out, P, zs, sst);
}